// FpsPool_33036888441076
// MI455X (gfx1250) — compile-verified
//
#include <hip/hip_runtime.h>

// ---------------------------------------------------------------------------
// FpsPool for MI455X (gfx1250, wave32).
//   K0: pack source points as (-2x,-2y,-2z,|p|^2) float4 (WMMA B operand).
//   K1: per-cloud FPS (serial argmax loop, block-parallel min/argmax)
//       + in-LDS bitonic sort of the 4096 sample indices.
//   K2: gather dst coords / batch ids / global dst indices.
//   K3: ball query: pairwise d2 via V_WMMA_F32_16X16X4_F32, 4 tiles/iter,
//       wave-vote-gated LDS candidate buckets + per-row top-K(32).
// ---------------------------------------------------------------------------

#define Bc   4
#define Nc   16384
#define Sc   4096          // Nc/4
#define Kc   32
#define CAP  64            // candidate buffer depth (E[count]~35, overflow ~never)

static __device__ constexpr float R2f  = 0.08f * 0.08f;
static __device__ constexpr float BIGF = 1e30f;

typedef __attribute__((ext_vector_type(2))) float v2f;
typedef __attribute__((ext_vector_type(8))) float v8f;

// ------------------------ K0: pack B operand (AoS float4) -------------------------
__global__ void pack_kernel(const float* __restrict__ pts,
                            float4* __restrict__ pack)
{
    const int g = blockIdx.x * blockDim.x + threadIdx.x;
    if (g >= Bc * Nc) return;
    const float x = pts[3 * g + 0];
    const float y = pts[3 * g + 1];
    const float z = pts[3 * g + 2];
    float4 q;
    q.x = -2.0f * x;
    q.y = -2.0f * y;
    q.z = -2.0f * z;
    q.w = x * x + y * y + z * z;
    pack[g] = q;
}

// ------------------------------ K1: FPS + sort -----------------------------
__global__ __launch_bounds__(1024) void fps_sort_kernel(
    const float* __restrict__ pts, int* __restrict__ fpsSorted)
{
    const int b = blockIdx.x;
    const int t = threadIdx.x;
    const int lane = t & 31;
    const int wave = t >> 5;
    const float* __restrict__ p = pts + (size_t)b * Nc * 3;

    __shared__ int   s_idx[Sc];       // FPS picks, later sorted (16 KB)
    __shared__ float s_wv[32];
    __shared__ int   s_wi[32];
    __shared__ int   s_win;

    // residual min-distance per owned point (16 points / thread, reg-resident)
    float dist[16];
#pragma unroll
    for (int j = 0; j < 16; ++j) dist[j] = BIGF;

    if (t == 0) s_idx[0] = 0;
    float lx = p[0], ly = p[1], lz = p[2];

    for (int it = 1; it < Sc; ++it) {
        float bv = -1.0f;
        int   bi = 0;
#pragma unroll
        for (int j = 0; j < 16; ++j) {
            const int i = j * 1024 + t;             // coalesced ownership
            const float dx = p[3 * i + 0] - lx;
            const float dy = p[3 * i + 1] - ly;
            const float dz = p[3 * i + 2] - lz;
            const float d  = dx * dx + dy * dy + dz * dz;
            const float nd = fminf(dist[j], d);
            dist[j] = nd;
            if (nd > bv) { bv = nd; bi = i; }       // strict > keeps lowest i
        }
        // wave32 argmax reduce (first-occurrence tie break = lowest index)
#pragma unroll
        for (int m = 16; m > 0; m >>= 1) {
            const float ov = __shfl_xor(bv, m, 32);
            const int   oi = __shfl_xor(bi, m, 32);
            if (ov > bv || (ov == bv && oi < bi)) { bv = ov; bi = oi; }
        }
        if (lane == 0) { s_wv[wave] = bv; s_wi[wave] = bi; }
        __syncthreads();
        if (wave == 0) {
            float v  = s_wv[lane];
            int   ix = s_wi[lane];
#pragma unroll
            for (int m = 16; m > 0; m >>= 1) {
                const float ov = __shfl_xor(v, m, 32);
                const int   oi = __shfl_xor(ix, m, 32);
                if (ov > v || (ov == v && oi < ix)) { v = ov; ix = oi; }
            }
            if (lane == 0) { s_win = ix; s_idx[it] = ix; }
        }
        __syncthreads();
        const int w = s_win;
        lx = p[3 * w + 0]; ly = p[3 * w + 1]; lz = p[3 * w + 2];
    }
    __syncthreads();

    // in-LDS bitonic sort of the 4096 picked indices (torch.unique == sort here)
    for (int k = 2; k <= Sc; k <<= 1) {
        for (int j = k >> 1; j > 0; j >>= 1) {
            __syncthreads();
            for (int i = t; i < Sc; i += 1024) {
                const int ixj = i ^ j;
                if (ixj > i) {
                    const int a = s_idx[i];
                    const int c = s_idx[ixj];
                    const bool up = ((i & k) == 0);
                    if ((a > c) == up) { s_idx[i] = c; s_idx[ixj] = a; }
                }
            }
        }
    }
    __syncthreads();
    for (int i = t; i < Sc; i += 1024) fpsSorted[b * Sc + i] = s_idx[i];
}

// ------------------------------ K2: gather meta ----------------------------
__global__ void meta_kernel(const float* __restrict__ pts,
                            const int* __restrict__ fpsSorted,
                            float* __restrict__ o_coord,
                            float* __restrict__ o_batch,
                            float* __restrict__ o_nidx)
{
    const int g = blockIdx.x * blockDim.x + threadIdx.x;
    if (g >= Bc * Sc) return;
    const int b   = g / Sc;
    const int idx = fpsSorted[g];
    const float* __restrict__ p = pts + ((size_t)b * Nc + idx) * 3;
    o_coord[3 * g + 0] = p[0];
    o_coord[3 * g + 1] = p[1];
    o_coord[3 * g + 2] = p[2];
    o_batch[g] = (float)b;
    o_nidx[g]  = (float)(idx + b * Nc);
}

// ---------------------- K3: WMMA ball query + top-K ------------------------
// One wave per 16-dst tile.  d2(i,j) = |a_i|^2 - 2 a_i.b_j + |b_j|^2 computed
// as D = A(16x4) x B(4x16) + C with A_i=(x,y,z,1), B_j=(-2bx,-2by,-2bz,|b|^2),
// C_ij=|a_i|^2  ->  exact f32 squared distances, 256 pairs per v_wmma.
// 4 src tiles per iteration; selection path gated by a wave vote so the
// common no-candidate case stays on the uniform WMMA/VMEM path.
__global__ __launch_bounds__(32) void ballq_kernel(
    const float* __restrict__ pts, const float4* __restrict__ pack,
    const int* __restrict__ fpsSorted,
    float* __restrict__ o_esrc, float* __restrict__ o_edst,
    float* __restrict__ o_mask, float* __restrict__ o_deg)
{
    const int blk  = blockIdx.x;            // Bc * (Sc/16) tiles
    const int b    = blk / (Sc / 16);
    const int tile = blk % (Sc / 16);
    const int lane = threadIdx.x;
    const int l16  = lane & 15;
    const bool hi  = lane >= 16;

    __shared__ float a2s[16];
    __shared__ int   fpsLoc[16];
    __shared__ int   cnt[16];
    __shared__ float candD[16][CAP];
    __shared__ int   candI[16][CAP];

    const float*  __restrict__ p  = pts  + (size_t)b * Nc * 3;
    const float4* __restrict__ pp = pack + (size_t)b * Nc;

    // dst points of this tile (each lane loads its own copy -> no shuffles)
    const int sLocal = tile * 16 + l16;
    const int fi = fpsSorted[b * Sc + sLocal];
    const float ax = p[3 * fi + 0], ay = p[3 * fi + 1], az = p[3 * fi + 2];
    const float a2 = ax * ax + ay * ay + az * az;
    if (!hi) { a2s[l16] = a2; fpsLoc[l16] = fi; cnt[l16] = 0; }
    __syncthreads();

    // A operand: lanes 0-15 hold (K=0,1)=(x,y) of row l16; lanes 16-31 (K=2,3)=(z,1)
    v2f A;
    A.x = hi ? az : ax;
    A.y = hi ? 1.0f : ay;

    // C operand: row (vgpr + 8*hi) broadcast of |a|^2
    v8f C0;
#pragma unroll
    for (int v = 0; v < 8; ++v) C0[v] = a2s[v + (hi ? 8 : 0)];

    for (int ts = 0; ts < Nc; ts += 64) {
        if (ts + 64 < Nc) __builtin_prefetch(&pp[ts + 64 + l16], 0, 0);

        v8f   D[4];
        float dmin[4];
#pragma unroll
        for (int u = 0; u < 4; ++u) {
            const float4 q = pp[ts + u * 16 + l16];
            // B operand: VGPR0 = rows 0,1 (-2bx | -2by), VGPR1 = rows 2,3 (-2bz | |b|^2)
            v2f Bm;
            Bm.x = hi ? q.y : q.x;
            Bm.y = hi ? q.w : q.z;
            D[u] = __builtin_amdgcn_wmma_f32_16x16x4_f32(
                /*neg_a=*/false, A, /*neg_b=*/false, Bm,
                /*c_mod=*/(short)0, C0, /*reuse_a=*/false, /*reuse_b=*/false);
            float dm = D[u][0];
#pragma unroll
            for (int v = 1; v < 8; ++v) dm = fminf(dm, D[u][v]);
            dmin[u] = dm;
        }

        // rare path: ~42% of 16-src tiles contain any in-ball pair
#pragma unroll
        for (int u = 0; u < 4; ++u) {
            if (__any(dmin[u] <= R2f)) {
                const int sp = ts + u * 16 + l16;
#pragma unroll
                for (int v = 0; v < 8; ++v) {
                    const float d = D[u][v];
                    if (d <= R2f) {             // self point (d~0) included, masked later
                        const int row = v + (hi ? 8 : 0);
                        const int pos = atomicAdd(&cnt[row], 1);
                        if (pos < CAP) { candD[row][pos] = d; candI[row][pos] = sp; }
                    }
                }
            }
        }
    }
    __syncthreads();

    // per-row exact top-K: sort <=64 candidates by (d2, idx), emit K slots
    if (lane < 16) {
        const int row = lane;
        int n = cnt[row];
        if (n > CAP) n = CAP;
        for (int i = 1; i < n; ++i) {
            const float dv = candD[row][i];
            const int   iv = candI[row][i];
            int j = i - 1;
            while (j >= 0 && (candD[row][j] > dv ||
                              (candD[row][j] == dv && candI[row][j] > iv))) {
                candD[row][j + 1] = candD[row][j];
                candI[row][j + 1] = candI[row][j];
                --j;
            }
            candD[row][j + 1] = dv;
            candI[row][j + 1] = iv;
        }
        const int sg   = tile * 16 + row;
        const int fpsI = fpsLoc[row];
        const long e   = (long)(b * Sc + sg) * Kc;
        const int  m   = n < Kc ? n : Kc;
        int deg = 0;
        for (int k = 0; k < Kc; ++k) {
            const int  srcLocal = (k < m) ? candI[row][k] : 0;
            const bool valid    = (k < m) && (srcLocal != fpsI);
            deg += valid ? 1 : 0;
            o_esrc[e + k] = (float)(srcLocal + b * Nc);
            o_edst[e + k] = (float)(b * Sc + sg);
            o_mask[e + k] = valid ? 1.0f : 0.0f;
        }
        o_deg[b * Sc + sg] = (float)deg;
    }
}

// --------------------------------- launch ----------------------------------
extern "C" void kernel_launch(void* const* d_in, const int* in_sizes, int n_in,
                              void* d_out, int out_size, void* d_ws, size_t ws_size,
                              hipStream_t stream)
{
    const float* pts = (const float*)d_in[0];      // [B*N, 3] f32
    // d_in[1] (batch_src) is implied by the fixed B,N layout; unused.

    float* out     = (float*)d_out;
    float* o_coord = out;                               // B*S*3
    float* o_esrc  = o_coord + (size_t)Bc * Sc * 3;     // B*S*K
    float* o_edst  = o_esrc  + (size_t)Bc * Sc * Kc;    // B*S*K
    float* o_mask  = o_edst  + (size_t)Bc * Sc * Kc;    // B*S*K
    float* o_deg   = o_mask  + (size_t)Bc * Sc * Kc;    // B*S
    float* o_batch = o_deg   + (size_t)Bc * Sc;         // B*S
    float* o_nidx  = o_batch + (size_t)Bc * Sc;         // B*S

    // scratch: [0,64KB) sorted FPS indices; [64KB, 64KB+1MB) packed B operand
    int*    fpsSorted = (int*)d_ws;
    float4* pack      = (float4*)((char*)d_ws + (size_t)Bc * Sc * sizeof(int));

    pack_kernel<<<(Bc * Nc + 255) / 256, 256, 0, stream>>>(pts, pack);
    fps_sort_kernel<<<Bc, 1024, 0, stream>>>(pts, fpsSorted);
    meta_kernel<<<(Bc * Sc + 255) / 256, 256, 0, stream>>>(
        pts, fpsSorted, o_coord, o_batch, o_nidx);
    ballq_kernel<<<Bc * (Sc / 16), 32, 0, stream>>>(
        pts, pack, fpsSorted, o_esrc, o_edst, o_mask, o_deg);
}